// ContrastiveLearningLoss_66116726554940
// MI455X (gfx1250) — compile-verified
//
#include <hip/hip_runtime.h>
#include <hip/hip_bf16.h>
#include <math.h>

// Problem constants (from reference)
#define MAXNUM 10
#define BB     4
#define CC     256
#define HH     64
#define WW     176
#define HWSZ   (HH * WW)          // 11264
#define NREG   (MAXNUM * BB)      // 40
#define NPAD   48                 // padded to 3x 16 tiles
#define TAUF   0.07f
#define EPSF   1e-12f

// Workspace layout (float offsets)
#define WS_SQ   0                        // 40*256 masked sums (q)
#define WS_SK   (WS_SQ + NREG * CC)      // 40*256 masked sums (k)
#define WS_CNT  (WS_SK + NREG * CC)      // 40 counts
#define WS_NK   (WS_CNT + 256)           // 48*256 normalized k (padded rows zero)
#define WS_NQ   (WS_NK + NPAD * CC)      // 48*256 normalized q
#define WS_PADF (WS_NQ + NPAD * CC)      // 48 pad flags
#define WS_TOTAL (WS_PADF + NPAD)

typedef __attribute__((ext_vector_type(2))) float v2f;
typedef __attribute__((ext_vector_type(8))) float v8f;

// ---------------------------------------------------------------------------
// Kernel 1: masked reduction over each (region, channel) plane for q and k.
// grid = NREG*CC blocks, 256 threads. Reads every feature element exactly once
// (float4 vectorized); mask plane re-read per channel comes from L2.
// ---------------------------------------------------------------------------
__global__ __launch_bounds__(256) void cl_reduce_kernel(
    const float* __restrict__ q, const float* __restrict__ k,
    const int* __restrict__ mask, float* __restrict__ ws) {
  const int nc = blockIdx.x;      // n*CC + c
  const int n  = nc >> 8;
  const int c  = nc & 255;
  const float4* __restrict__ qp = (const float4*)(q + (size_t)nc * HWSZ);
  const float4* __restrict__ kp = (const float4*)(k + (size_t)nc * HWSZ);
  const int4*   __restrict__ mp = (const int4*)(mask + (size_t)n * HWSZ);

  float sq = 0.f, sk = 0.f, cm = 0.f;
  for (int i = threadIdx.x; i < HWSZ / 4; i += 256) {
    float4 qv = qp[i];
    float4 kv = kp[i];
    int4   mv = mp[i];
    float m0 = mv.x ? 1.f : 0.f;
    float m1 = mv.y ? 1.f : 0.f;
    float m2 = mv.z ? 1.f : 0.f;
    float m3 = mv.w ? 1.f : 0.f;
    sq += qv.x * m0 + qv.y * m1 + qv.z * m2 + qv.w * m3;
    sk += kv.x * m0 + kv.y * m1 + kv.z * m2 + kv.w * m3;
    cm += m0 + m1 + m2 + m3;
  }

  __shared__ float rq[256], rk[256], rc[256];
  const int t = threadIdx.x;
  rq[t] = sq; rk[t] = sk; rc[t] = cm;
  __syncthreads();
  for (int s = 128; s > 0; s >>= 1) {
    if (t < s) { rq[t] += rq[t + s]; rk[t] += rk[t + s]; rc[t] += rc[t + s]; }
    __syncthreads();
  }
  if (t == 0) {
    ws[WS_SQ + nc] = rq[0];
    ws[WS_SK + nc] = rk[0];
    if (c == 0) ws[WS_CNT + n] = rc[0];
  }
}

// ---------------------------------------------------------------------------
// Kernel 2: means, L2 norms (eps-clamped), pad flags; write zero-padded
// 48x256 normalized matrices for the WMMA GEMM. Single block, 256 threads.
// ---------------------------------------------------------------------------
__global__ __launch_bounds__(256) void cl_normalize_kernel(float* __restrict__ ws) {
  __shared__ float s_invq[NREG], s_invk[NREG], s_pad[NREG];
  const int t = threadIdx.x;
  if (t < NREG) {
    float cnt = fmaxf(ws[WS_CNT + t], 1.f);
    float ic  = 1.f / cnt;
    float aq = 0.f, ak = 0.f;
    for (int c = 0; c < CC; ++c) {
      float mq = ws[WS_SQ + t * CC + c] * ic;
      float mk = ws[WS_SK + t * CC + c] * ic;
      aq += mq * mq;
      ak += mk * mk;
    }
    s_invq[t] = 1.f / fmaxf(sqrtf(aq), EPSF);
    s_invk[t] = 1.f / fmaxf(sqrtf(ak), EPSF);
    s_pad[t]  = ((ws[WS_SK + t * CC] * ic) != 0.f) ? 1.f : 0.f;
  }
  __syncthreads();
  for (int idx = t; idx < NPAD * CC; idx += 256) {
    int r = idx >> 8, c = idx & 255;
    float vq = 0.f, vk = 0.f;
    if (r < NREG) {
      float ic = 1.f / fmaxf(ws[WS_CNT + r], 1.f);
      vq = ws[WS_SQ + r * CC + c] * ic * s_invq[r];
      vk = ws[WS_SK + r * CC + c] * ic * s_invk[r];
    }
    ws[WS_NQ + idx] = vq;
    ws[WS_NK + idx] = vk;
  }
  if (t < NPAD) ws[WS_PADF + t] = (t < NREG) ? s_pad[t] : 0.f;
}

// ---------------------------------------------------------------------------
// Kernel 3: sim = nk @ nq^T (48x48 padded) via V_WMMA_F32_16X16X4_F32,
// then row log-softmax, diagonal CE, pad-masked mean. Single block, 9 waves.
// A layout (16x4 f32): lane l -> M=l%16, holds K = 2*(l/16), 2*(l/16)+1.
// B layout symmetric: lane l -> N=l%16, same K pair.
// C/D layout: VGPR v -> row v + 8*(l/16), col l%16.
// ---------------------------------------------------------------------------
__global__ __launch_bounds__(288) void cl_simloss_kernel(float* __restrict__ ws,
                                                         float* __restrict__ out) {
  __shared__ float sim[NPAD][NPAD];
  __shared__ float s_ce[NREG];

  const int tid  = threadIdx.x;
  const int wave = tid >> 5;
  const int lane = tid & 31;
  const int ti = wave / 3;              // row tile (0..2)
  const int tj = wave % 3;              // col tile (0..2)
  const int mrow  = ti * 16 + (lane & 15);
  const int ncol  = tj * 16 + (lane & 15);
  const int khalf = (lane >> 4) * 2;    // 0 or 2

  const float* __restrict__ nk = ws + WS_NK;
  const float* __restrict__ nq = ws + WS_NQ;

  v8f acc = {0.f, 0.f, 0.f, 0.f, 0.f, 0.f, 0.f, 0.f};
  for (int kb = 0; kb < CC; kb += 4) {
    v2f a, b;
    a.x = nk[mrow * CC + kb + khalf];
    a.y = nk[mrow * CC + kb + khalf + 1];
    b.x = nq[ncol * CC + kb + khalf];
    b.y = nq[ncol * CC + kb + khalf + 1];
    acc = __builtin_amdgcn_wmma_f32_16x16x4_f32(
        /*neg_a=*/false, a, /*neg_b=*/false, b,
        /*c_mod=*/(short)0, acc, /*reuse_a=*/false, /*reuse_b=*/false);
  }

  const int rbase = ti * 16 + 8 * (lane >> 4);
#pragma unroll
  for (int v = 0; v < 8; ++v) {
    sim[rbase + v][ncol] = acc[v];
  }
  __syncthreads();

  if (tid < NREG) {
    float mx = -1e30f;
    for (int j = 0; j < NREG; ++j) mx = fmaxf(mx, sim[tid][j] / TAUF);
    float se = 0.f;
    for (int j = 0; j < NREG; ++j) se += expf(sim[tid][j] / TAUF - mx);
    float logp_diag = sim[tid][tid] / TAUF - mx - logf(se);
    s_ce[tid] = -logp_diag * ws[WS_PADF + tid];
  }
  __syncthreads();

  if (tid == 0) {
    float num = 0.f, den = 0.f;
    for (int n = 0; n < NREG; ++n) {
      num += s_ce[n];
      den += ws[WS_PADF + n];
    }
    out[0] = num / den;
  }
}

// ---------------------------------------------------------------------------
extern "C" void kernel_launch(void* const* d_in, const int* in_sizes, int n_in,
                              void* d_out, int out_size, void* d_ws, size_t ws_size,
                              hipStream_t stream) {
  const float* q    = (const float*)d_in[0];
  const float* k    = (const float*)d_in[1];
  const int*   mask = (const int*)d_in[2];
  float* ws  = (float*)d_ws;
  float* out = (float*)d_out;

  cl_reduce_kernel<<<NREG * CC, 256, 0, stream>>>(q, k, mask, ws);
  cl_normalize_kernel<<<1, 256, 0, stream>>>(ws);
  cl_simloss_kernel<<<1, 288, 0, stream>>>(ws, out);
}